// AttentionLayer_29059748725397
// MI455X (gfx1250) — compile-verified
//
#include <hip/hip_runtime.h>
#include <math.h>

#define EMBED 128
#define ATT   64
#define SEQ   200
#define MTILES 13          // ceil(200/16)
#define THREADS 256
#define NWAVES (THREADS/32)

typedef __attribute__((ext_vector_type(2))) float v2f;
typedef __attribute__((ext_vector_type(8))) float v8f;

#if defined(__has_builtin) && __has_builtin(__builtin_amdgcn_tanhf)
#define FAST_TANH(x) __builtin_amdgcn_tanhf(x)      // native V_TANH_F32
#else
#define FAST_TANH(x) tanhf(x)
#endif

__launch_bounds__(THREADS)
__global__ void attn_addatt_kernel(const float* __restrict__ query,
                                   const float* __restrict__ keys,
                                   const float* __restrict__ values,
                                   const int*   __restrict__ mask,
                                   const float* __restrict__ W_q,
                                   const float* __restrict__ b_q,
                                   const float* __restrict__ W_k,
                                   const float* __restrict__ b_k,
                                   const float* __restrict__ w_e,
                                   float* __restrict__ out)
{
    // W_k pre-swizzled into K-pair-major layout: element (k, a) lives at
    // sWkP[(k>>1)*128 + 2*a + (k&1)], so the B fragment {W_k[k][a], W_k[k+1][a]}
    // is one contiguous, 8B-aligned float2 -> single ds_load_b64, no v_movs.
    __shared__ float sWkP[EMBED * ATT];    // 32 KB
    __shared__ float sC[ATT];              // q_proj + b_q + b_k
    __shared__ float sWe[ATT];
    __shared__ float sScore[MTILES * 16];  // 208 scores -> attn weights
    __shared__ float sRed[4 * ATT];        // q_proj partial reduction
    __shared__ float2 sCtx[THREADS];       // context partials (float2 each)

    const int b    = blockIdx.x;
    const int t    = threadIdx.x;
    const int lane = t & 31;
    const int wave = t >> 5;

    const float* keysb = keys   + (size_t)b * SEQ * EMBED;
    const float* valb  = values + (size_t)b * SEQ * EMBED;

    // ---- Stage W_k into LDS, swizzled (coalesced global reads) ----
    for (int i = t; i < EMBED * ATT; i += THREADS) {
        const int k = i >> 6;              // row (K index)
        const int a = i & (ATT - 1);       // col (A index)
        sWkP[(k >> 1) * 128 + 2 * a + (k & 1)] = W_k[i];
    }

    // ---- Prefetch the values tile (100 KB = 800 lines) toward L2 while we
    //      do the WMMA phase; phase 3 consumes it after the softmax barrier.
    {
        const char* vb = (const char*)valb;
        #pragma unroll
        for (int i = 0; i < 4; ++i) {
            const int line = t + i * THREADS;      // 0..1023, need 800
            if (line < (SEQ * EMBED * 4) / 128)
                __builtin_prefetch(vb + (size_t)line * 128, 0, 1);
        }
    }

    // ---- q_proj = query[b] @ W_q  (4 groups of 32 k's, reduced via LDS) ----
    {
        const int a = t & (ATT - 1);
        const int g = t >> 6;                       // 0..3
        const float* qrow = query + (size_t)b * EMBED + g * 32;
        float p = 0.f;
        #pragma unroll
        for (int k = 0; k < 32; ++k)
            p = fmaf(qrow[k], W_q[(size_t)(g * 32 + k) * ATT + a], p);
        sRed[g * ATT + a] = p;
    }
    __syncthreads();
    if (t < ATT) {
        sC[t]  = sRed[t] + sRed[ATT + t] + sRed[2 * ATT + t] + sRed[3 * ATT + t]
               + b_q[t] + b_k[t];
        sWe[t] = w_e[t];
    }
    __syncthreads();

    // ---- Phase 1: k_proj = keys[b] @ W_k via V_WMMA_F32_16X16X4_F32.
    //      kc outer / nt inner: A chunk loaded once, 4 live accumulators.
    //      Fused epilogue: native tanh(. + c)·w_e, 16-lane reduce -> scores.
    for (int mt = wave; mt < MTILES; mt += NWAVES) {
        const int mbase = mt * 16;
        const int row   = mbase + (lane & 15);      // A row this lane supplies
        const int kofs  = (lane >> 4) * 2;          // K sub-offset per ISA A layout
        const int rowc  = (row < SEQ) ? row : (SEQ - 1);   // clamped, always valid
        const float am  = (row < SEQ) ? 1.f : 0.f;         // zero pad rows >= SEQ
        const v2f* kp2  = (const v2f*)(keysb + (size_t)rowc * EMBED + kofs);
        const int nc    = lane & 15;
        // B fragment base (float2 index): pair j = (kc*4+kofs)/2 = 2*kc + kofs/2
        const v2f* bbase = (const v2f*)sWkP + (kofs >> 1) * 64 + nc;

        v8f acc[4];
        #pragma unroll
        for (int nt = 0; nt < 4; ++nt)
            acc[nt] = (v8f){0.f, 0.f, 0.f, 0.f, 0.f, 0.f, 0.f, 0.f};

        #pragma unroll 4
        for (int kc = 0; kc < 32; ++kc) {           // K = 128 in steps of 4
            v2f a = kp2[kc * 2];                    // {keys[row][4kc+kofs], +1}
            a.x *= am;
            a.y *= am;
            const v2f* bp = bbase + (size_t)kc * 128;   // pair row 2*kc
            #pragma unroll
            for (int nt = 0; nt < 4; ++nt) {
                const v2f bf = bp[nt * 16];
                acc[nt] = __builtin_amdgcn_wmma_f32_16x16x4_f32(
                            false, a, false, bf, (short)0, acc[nt], false, false);
            }
        }

        // Epilogue: D layout lane L, vgpr v -> M = v + 8*(L>=16), N = L&15
        float sacc[8] = {0.f, 0.f, 0.f, 0.f, 0.f, 0.f, 0.f, 0.f};
        #pragma unroll
        for (int nt = 0; nt < 4; ++nt) {
            const int ncol   = nt * 16 + nc;
            const float cadd = sC[ncol];
            const float wev  = sWe[ncol];
            #pragma unroll
            for (int v = 0; v < 8; ++v)
                sacc[v] += FAST_TANH(acc[nt][v] + cadd) * wev;
        }

        // Reduce the 16 N-lanes of each half-wave -> per-row score
        #pragma unroll
        for (int v = 0; v < 8; ++v) {
            float r = sacc[v];
            r += __shfl_xor(r, 1, 32);
            r += __shfl_xor(r, 2, 32);
            r += __shfl_xor(r, 4, 32);
            r += __shfl_xor(r, 8, 32);
            if ((lane & 15) == 0)
                sScore[mbase + v + 8 * (lane >> 4)] = r;
        }
    }
    __syncthreads();

    // ---- Phase 2: masked softmax over S=200 (wave 0) ----
    if (wave == 0) {
        const int* mrow = mask + (size_t)b * SEQ;
        float sc[7];
        float lmax = -3.0e38f;
        #pragma unroll
        for (int i = 0; i < 7; ++i) {
            const int s = lane + i * 32;
            float v = -3.0e38f;
            if (s < SEQ) {
                v = sScore[s];
                if (mrow[s] == 0) v = -1.0e9f;
            }
            sc[i] = v;
            lmax = fmaxf(lmax, v);
        }
        #pragma unroll
        for (int off = 16; off >= 1; off >>= 1)
            lmax = fmaxf(lmax, __shfl_xor(lmax, off, 32));
        float lsum = 0.f;
        #pragma unroll
        for (int i = 0; i < 7; ++i) {
            const int s = lane + i * 32;
            if (s < SEQ) {
                const float e = __expf(sc[i] - lmax);
                sScore[s] = e;
                lsum += e;
            }
        }
        #pragma unroll
        for (int off = 16; off >= 1; off >>= 1)
            lsum += __shfl_xor(lsum, off, 32);
        const float inv = 1.0f / lsum;
        #pragma unroll
        for (int i = 0; i < 7; ++i) {
            const int s = lane + i * 32;
            if (s < SEQ) sScore[s] *= inv;
        }
    }
    __syncthreads();

    // ---- Phase 3: context = attn @ values[b]  (float2, 4-way S split) ----
    {
        const float2* valb2 = (const float2*)valb;  // 64 float2 per row
        const int e2 = t & 63;
        const int q  = t >> 6;                      // 0..3, 50 rows each
        float2 ctx; ctx.x = 0.f; ctx.y = 0.f;
        const int s0 = q * (SEQ / 4);
        for (int s = s0; s < s0 + SEQ / 4; ++s) {
            const float  w = sScore[s];
            const float2 v = valb2[(size_t)s * (EMBED / 2) + e2];
            ctx.x = fmaf(w, v.x, ctx.x);
            ctx.y = fmaf(w, v.y, ctx.y);
        }
        sCtx[t] = ctx;
    }
    __syncthreads();
    if (t < EMBED / 2) {
        float2 r;
        r.x = sCtx[t].x + sCtx[t + 64].x + sCtx[t + 128].x + sCtx[t + 192].x;
        r.y = sCtx[t].y + sCtx[t + 64].y + sCtx[t + 128].y + sCtx[t + 192].y;
        out[(size_t)b * EMBED + 2 * t + 0] = r.x;
        out[(size_t)b * EMBED + 2 * t + 1] = r.y;
    }
}

extern "C" void kernel_launch(void* const* d_in, const int* in_sizes, int n_in,
                              void* d_out, int out_size, void* d_ws, size_t ws_size,
                              hipStream_t stream) {
    const float* query  = (const float*)d_in[0];
    const float* keys   = (const float*)d_in[1];
    const float* values = (const float*)d_in[2];
    const int*   mask   = (const int*)  d_in[3];
    const float* W_q    = (const float*)d_in[4];
    const float* b_q    = (const float*)d_in[5];
    const float* W_k    = (const float*)d_in[6];
    const float* b_k    = (const float*)d_in[7];
    const float* w_e    = (const float*)d_in[8];
    float* out = (float*)d_out;

    const int B = in_sizes[0] / EMBED;   // 4096

    attn_addatt_kernel<<<B, THREADS, 0, stream>>>(
        query, keys, values, mask, W_q, b_q, W_k, b_k, w_e, out);
}